// ACAT_29154238005975
// MI455X (gfx1250) — compile-verified
//
#include <hip/hip_runtime.h>

// MI455X / gfx1250, wave32. Full-precision WMMA (V_WMMA_F32_16X16X4_F32) because
// score magnitudes (~2e3 std) make softmax intolerant of 16-bit matmul error.

typedef float v2f __attribute__((ext_vector_type(2)));
typedef float v8f __attribute__((ext_vector_type(8)));

#define L_SEQ 2048
#define D_MODEL 512
#define D_K 64
#define NB 4
#define NH 8

__device__ __forceinline__ v8f wmma4(v2f a, v2f b, v8f c) {
  // 8 args: (neg_a, A, neg_b, B, c_mod, C, reuse_a, reuse_b)
  return __builtin_amdgcn_wmma_f32_16x16x4_f32(false, a, false, b, (short)0, c,
                                               false, false);
}

// ---------------------------------------------------------------------------
// Conv stage: Out[b, t*512+o] = X[b, t*512+o] +
//   sum_{c<512, f<nf} X[b, c*2048 + (t+f-2)] * W[(c*512+o)*4 + f]
// One wave computes one 16(t) x 16(o) tile; block = 4 waves = 64 o-columns.
// Boundary handling is branch-free: clamped address + v_cndmask zero-select,
// so the inner loop is pure load -> v_wmma with no EXEC-mask regions.
// ---------------------------------------------------------------------------
__global__ __launch_bounds__(128) void conv_qk_kernel(
    const float* __restrict__ X, const float* __restrict__ W,
    const float* __restrict__ wsel, float* __restrict__ Out) {
  const int lane = threadIdx.x & 31;
  const int wid = threadIdx.x >> 5;
  const int half = lane >> 4;
  const int ln = lane & 15;
  const int tbase = blockIdx.y * 16;
  const int obase = blockIdx.x * 64 + wid * 16;
  const float* Xb = X + (size_t)blockIdx.z * (D_MODEL * L_SEQ);
  float* Ob = Out + (size_t)blockIdx.z * (D_MODEL * L_SEQ);

  // chosen_len = FILTER_LENGTHS[argmax([2*w0, 4*w1])]; argmax picks first on tie
  const float w0 = wsel[0], w1 = wsel[1];
  const int nf = (2.0f * w0 >= 4.0f * w1) ? 2 : 4;

  v8f acc;
  for (int i = 0; i < 8; ++i) acc[i] = 0.0f;

  for (int f = 0; f < nf; ++f) {
    const int tpos = tbase + ln + f - 2;  // A row m = ln
    const bool inb = (tpos >= 0) && (tpos < L_SEQ);
    const int tcl = min(max(tpos, 0), L_SEQ - 1);  // always-legal address
    const float* xa = Xb + tcl;
    const float* wp = W + (size_t)(obase + ln) * 4 + f;
#pragma unroll 4
    for (int c0 = 0; c0 < D_MODEL; c0 += 4) {
      const int cA = c0 + 2 * half;  // K-pair base for this lane half
      float va0 = xa[(size_t)cA * L_SEQ];
      float va1 = xa[(size_t)(cA + 1) * L_SEQ];
      v2f a, b;
      a.x = inb ? va0 : 0.0f;  // v_cndmask, no divergent branch
      a.y = inb ? va1 : 0.0f;
      b.x = wp[(size_t)cA * (D_MODEL * 4)];
      b.y = wp[(size_t)(cA + 1) * (D_MODEL * 4)];
      acc = wmma4(a, b, acc);
    }
  }
#pragma unroll
  for (int v = 0; v < 8; ++v) {
    const int t = tbase + v + 8 * half;
    const int idx = t * D_MODEL + obase + ln;
    Ob[idx] = acc[v] + Xb[idx];  // residual add (same flat layout)
  }
}

// ---------------------------------------------------------------------------
// Fused attention: per block, a 16-row q-strip of one (b,h).
// 8 waves; wave w owns k in [w*256, w*256+256): 16 score tiles in registers.
// scores -> block softmax -> write attn once -> PV WMMA -> block reduce ctx.
// All contiguous fragment pairs loaded as v2f -> global_load_b64/ds_load_b64.
// ---------------------------------------------------------------------------
__global__ __launch_bounds__(256) void attn_kernel(
    const float* __restrict__ Qc, const float* __restrict__ Kc,
    const float* __restrict__ V, float* __restrict__ ctx,
    float* __restrict__ attn) {
  __shared__ float red[8 * 16];        // per-wave row max / sum
  __shared__ float ptile[8][16 * 16];  // per-wave P-tile transpose staging
  __shared__ float cred[8][16 * 64];   // per-wave partial context

  const int lane = threadIdx.x & 31;
  const int wid = threadIdx.x >> 5;
  const int half = lane >> 4;
  const int ln = lane & 15;
  const int bh = blockIdx.y;
  const int q0 = blockIdx.x * 16;
  const int kw0 = wid * 256;

  const float* Qb = Qc + (size_t)bh * (L_SEQ * D_K);
  const float* Kb = Kc + (size_t)bh * (L_SEQ * D_K);
  const float* Vb = V + (size_t)bh * (L_SEQ * D_K);

  // Preload Q A-fragments for all 16 d-steps (K=64 in steps of 4)
  v2f afrag[16];
  {
    const float* qp = Qb + (size_t)(q0 + ln) * D_K + 2 * half;
#pragma unroll
    for (int s = 0; s < 16; ++s)
      afrag[s] = *(const v2f*)(qp + s * 4);  // 8B aligned -> b64 load
  }

  // -------- scores: S = Qc * Kc^T / 8, kept in registers ------------------
  v8f sacc[16];
#pragma unroll
  for (int t = 0; t < 16; ++t) {
    v8f c;
    for (int i = 0; i < 8; ++i) c[i] = 0.0f;
    const float* Krow = Kb + (size_t)(kw0 + t * 16 + ln) * D_K + 2 * half;
#pragma unroll
    for (int s = 0; s < 16; ++s) {
      v2f b = *(const v2f*)(Krow + s * 4);  // b64 load
      c = wmma4(afrag[s], b, c);
    }
    for (int i = 0; i < 8; ++i) sacc[t][i] = c[i] * 0.125f;  // 1/sqrt(64)
  }

  // -------- row max (intra-wave shuffle, cross-wave via LDS) --------------
  float rmax[8];
#pragma unroll
  for (int v = 0; v < 8; ++v) {
    float m = sacc[0][v];
    for (int t = 1; t < 16; ++t) m = fmaxf(m, sacc[t][v]);
    for (int off = 1; off < 16; off <<= 1) m = fmaxf(m, __shfl_xor(m, off));
    rmax[v] = m;
  }
  if (ln == 0)
    for (int v = 0; v < 8; ++v) red[wid * 16 + half * 8 + v] = rmax[v];
  __syncthreads();
#pragma unroll
  for (int v = 0; v < 8; ++v) {
    float m = red[half * 8 + v];
    for (int w2 = 1; w2 < 8; ++w2) m = fmaxf(m, red[w2 * 16 + half * 8 + v]);
    rmax[v] = m;
  }
  __syncthreads();

  // -------- exp + row sum -------------------------------------------------
  float rsum[8];
#pragma unroll
  for (int v = 0; v < 8; ++v) {
    float s = 0.0f;
    for (int t = 0; t < 16; ++t) {
      float p = __expf(sacc[t][v] - rmax[v]);
      sacc[t][v] = p;
      s += p;
    }
    for (int off = 1; off < 16; off <<= 1) s += __shfl_xor(s, off);
    rsum[v] = s;
  }
  if (ln == 0)
    for (int v = 0; v < 8; ++v) red[wid * 16 + half * 8 + v] = rsum[v];
  __syncthreads();
#pragma unroll
  for (int v = 0; v < 8; ++v) {
    float s = 0.0f;
    for (int w2 = 0; w2 < 8; ++w2) s += red[w2 * 16 + half * 8 + v];
    rmax[v] = 1.0f / s;  // reuse as reciprocal of denominator
  }
  __syncthreads();

  // -------- normalize, write attn exactly once ----------------------------
  float* ab = attn + (size_t)bh * L_SEQ * L_SEQ;
#pragma unroll
  for (int t = 0; t < 16; ++t) {
    const int kc = kw0 + t * 16 + ln;
#pragma unroll
    for (int v = 0; v < 8; ++v) {
      float p = sacc[t][v] * rmax[v];
      sacc[t][v] = p;
      ab[(size_t)(q0 + v + 8 * half) * L_SEQ + kc] = p;
    }
  }

  // -------- context: ctx_partial = P(16x256) * V(256x64) ------------------
  v8f cacc[4];
  for (int n = 0; n < 4; ++n)
    for (int i = 0; i < 8; ++i) cacc[n][i] = 0.0f;

  for (int t = 0; t < 16; ++t) {
    float* my = ptile[wid];  // transpose C-layout tile -> A-layout via LDS
#pragma unroll
    for (int v = 0; v < 8; ++v) my[(v + 8 * half) * 16 + ln] = sacc[t][v];
    __syncthreads();
    const int kb = kw0 + t * 16;
#pragma unroll
    for (int s = 0; s < 4; ++s) {
      const int kk = s * 4 + 2 * half;
      v2f a = *(const v2f*)(my + ln * 16 + kk);  // ds_load_b64 (kk even)
#pragma unroll
      for (int n = 0; n < 4; ++n) {
        const float* vp = Vb + (size_t)(kb + kk) * D_K + n * 16 + ln;
        v2f b;
        b.x = vp[0];
        b.y = vp[D_K];
        cacc[n] = wmma4(a, b, cacc[n]);
      }
    }
    __syncthreads();
  }

  // -------- cross-wave sum of partial contexts ----------------------------
#pragma unroll
  for (int n = 0; n < 4; ++n)
    for (int v = 0; v < 8; ++v)
      cred[wid][(v + 8 * half) * 64 + n * 16 + ln] = cacc[n][v];
  __syncthreads();
  float* cb = ctx + (size_t)bh * (L_SEQ * D_K) + (size_t)q0 * D_K;
  for (int e = threadIdx.x; e < 16 * 64; e += 256) {
    float s = 0.0f;
    for (int w2 = 0; w2 < 8; ++w2) s += cred[w2][e];
    cb[e] = s;
  }
}

// ---------------------------------------------------------------------------
extern "C" void kernel_launch(void* const* d_in, const int* in_sizes, int n_in,
                              void* d_out, int out_size, void* d_ws,
                              size_t ws_size, hipStream_t stream) {
  const float* Q = (const float*)d_in[0];
  const float* K = (const float*)d_in[1];
  const float* V = (const float*)d_in[2];
  // d_in[3] = attn_mask (all-false, unused by the reference math)
  const float* conv_q = (const float*)d_in[4];
  const float* conv_k = (const float*)d_in[5];
  const float* w = (const float*)d_in[6];

  float* ctx = (float*)d_out;  // (4,8,2048,64) first
  float* attn = (float*)d_out + (size_t)NB * NH * L_SEQ * D_K;

  // workspace: Qc, Kc each NB*512*2048 floats (16.8 MB each)
  float* Qc = (float*)d_ws;
  float* Kc = Qc + (size_t)NB * D_MODEL * L_SEQ;

  dim3 cgrid(D_MODEL / 64, L_SEQ / 16, NB);
  conv_qk_kernel<<<cgrid, 128, 0, stream>>>(Q, conv_q, w, Qc);
  conv_qk_kernel<<<cgrid, 128, 0, stream>>>(K, conv_k, w, Kc);

  dim3 agrid(L_SEQ / 16, NB * NH);
  attn_kernel<<<agrid, 256, 0, stream>>>(Qc, Kc, V, ctx, attn);
}